// RankingLoss_43963285241920
// MI455X (gfx1250) — compile-verified
//
#include <hip/hip_runtime.h>
#include <hip/hip_bf16.h>

typedef __attribute__((ext_vector_type(2))) float v2f;
typedef __attribute__((ext_vector_type(8))) float v8f;

#define GAMMA_ 0.1f
#define C_     0.0f
#define B_     16
#define L_     2048
#define LT_    (L_ / 16)              // 128 tiles per dimension
#define WAVES_PER_BLOCK 8             // 256 threads, wave32

__global__ void RankingLoss_zero_kernel(float* out) {
    if (threadIdx.x == 0 && blockIdx.x == 0) out[0] = 0.0f;
}

__device__ __forceinline__ float fast_sigmoid(float v) {
    // v_exp_f32 + v_rcp_f32 (avoid IEEE divide expansion)
    return __builtin_amdgcn_rcpf(1.0f + __expf(-v));
}

__global__ __launch_bounds__(256) void RankingLoss_43963285241920_kernel(
    const float* __restrict__ input,   // [B, L]
    const int*   __restrict__ target,  // [B, L]
    const float* __restrict__ freq,    // [L]
    float*       __restrict__ out)     // [1]
{
    __shared__ float blockSum;

    const int lane   = threadIdx.x & 31;
    const int waveId = blockIdx.x * WAVES_PER_BLOCK + (threadIdx.x >> 5);
    const int I = waveId / LT_;        // row tile (pos side 'a')
    const int J = waveId % LT_;        // col tile (other side 'j')

    if (threadIdx.x == 0) blockSum = 0.0f;
    __syncthreads();

    const int  m  = lane & 15;         // M for A-side, N for B/C/D-side
    const bool hi = (lane >= 16);      // lanes 16..31 carry K=2/K=3 slots
    const int  rA = I * 16 + m;        // global row index a
    const int  cJ = J * 16 + m;        // global col index j

    // ---- batch-invariant frequency tile: n[a,j] = (fa - fj)/(fa + fj) ----
    // A_f rows: [fa, 1, 0, 0]; B_S cols: [1; fj; 0; 0]; B_D cols: [1; -fj; 0; 0]
    const float fa = freq[rA];
    const float fj = freq[cJ];

    v2f Af, Bs, Bd;
    Af.x = hi ? 0.0f : fa;    Af.y = hi ? 0.0f : 1.0f;
    Bs.x = hi ? 0.0f : 1.0f;  Bs.y = hi ? 0.0f : fj;
    Bd.x = hi ? 0.0f : 1.0f;  Bd.y = hi ? 0.0f : -fj;

    v8f zero = {};
    v8f S = __builtin_amdgcn_wmma_f32_16x16x4_f32(false, Af, false, Bs,
                                                  (short)0, zero, false, false);
    v8f D = __builtin_amdgcn_wmma_f32_16x16x4_f32(false, Af, false, Bd,
                                                  (short)0, zero, false, false);
    float nfac[8];
#pragma unroll
    for (int k = 0; k < 8; ++k) nfac[k] = D[k] * __builtin_amdgcn_rcpf(S[k]);

    // ---- per-batch masked pairwise terms via rank-<=3 WMMAs ----
    v8f acc = {};
#pragma unroll 4
    for (int b = 0; b < B_; ++b) {
        const float xa = fast_sigmoid(input[b * L_ + rA]);
        const float xj = fast_sigmoid(input[b * L_ + cJ]);
        const float pa = (float)target[b * L_ + rA];
        const float pj = (float)target[b * L_ + cJ];
        const float nj = 1.0f - pj;

        // term1 arg: pp*(xa - xj) = [pa*xa, pa] . [pj; -pj*xj]
        v2f A1, B1;
        A1.x = hi ? 0.0f : pa * xa;  A1.y = hi ? 0.0f : pa;
        B1.x = hi ? 0.0f : pj;       B1.y = hi ? 0.0f : -pj * xj;

        // term2 arg: pa*nj*(C + xj - xa) = [pa*C, pa, -pa*xa] . [nj; nj*xj; nj]
        // K=2 slot rides in lanes 16..31 of VGPR0 (same row m as low lanes).
        v2f A2, B2;
        A2.x = hi ? (-pa * xa) : (pa * C_);  A2.y = hi ? 0.0f : pa;
        B2.x = nj;                            B2.y = hi ? 0.0f : nj * xj;

        v8f U1 = __builtin_amdgcn_wmma_f32_16x16x4_f32(false, A1, false, B1,
                                                       (short)0, zero, false, false);
        v8f U2 = __builtin_amdgcn_wmma_f32_16x16x4_f32(false, A2, false, B2,
                                                       (short)0, zero, false, false);
#pragma unroll
        for (int k = 0; k < 8; ++k) {
            // masks pulled inside relu (masks are nonnegative)
            acc[k] += fmaxf(nfac[k] * U1[k], 0.0f) + GAMMA_ * fmaxf(U2[k], 0.0f);
        }
    }

    // ---- reduce: 8 accumulators -> lane -> wave32 shuffle -> LDS -> 1 atomic/block
    float s = 0.0f;
#pragma unroll
    for (int k = 0; k < 8; ++k) s += acc[k];
#pragma unroll
    for (int off = 16; off >= 1; off >>= 1) s += __shfl_xor(s, off, 32);

    if (lane == 0) atomicAdd(&blockSum, s);   // ds_add_f32, 8 per block
    __syncthreads();
    if (threadIdx.x == 0)
        atomicAdd(out, blockSum * (1.0f / (float)B_));  // 2048 global atomics total
}

extern "C" void kernel_launch(void* const* d_in, const int* in_sizes, int n_in,
                              void* d_out, int out_size, void* d_ws, size_t ws_size,
                              hipStream_t stream) {
    (void)in_sizes; (void)n_in; (void)out_size; (void)d_ws; (void)ws_size;
    const float* input  = (const float*)d_in[0];
    const int*   target = (const int*)d_in[1];
    const float* freq   = (const float*)d_in[2];
    float*       out    = (float*)d_out;

    RankingLoss_zero_kernel<<<1, 32, 0, stream>>>(out);

    const int tilePairs = LT_ * LT_;                    // 16384 waves of work
    const int blocks    = tilePairs / WAVES_PER_BLOCK;  // 2048 blocks x 256 thr
    RankingLoss_43963285241920_kernel<<<blocks, 256, 0, stream>>>(
        input, target, freq, out);
}